// NodeAttention_10685878632691
// MI455X (gfx1250) — compile-verified
//
#include <hip/hip_runtime.h>

#define DF 128
#define SLOPE 0.2f

typedef __attribute__((ext_vector_type(2))) float v2f;
typedef __attribute__((ext_vector_type(4))) float v4f;
typedef __attribute__((ext_vector_type(8))) float v8f;

// ---------------- device-scope float atomics (native CDNA5 L2 atomics) ------
__device__ __forceinline__ void atom_add_f32(float* p, float v) {
    asm volatile("global_atomic_add_f32 %0, %1, off scope:SCOPE_DEV"
                 :: "v"(p), "v"(v) : "memory");
}
__device__ __forceinline__ void atom_max_f32(float* p, float v) {
    asm volatile("global_atomic_max_num_f32 %0, %1, off scope:SCOPE_DEV"
                 :: "v"(p), "v"(v) : "memory");
}

// ---------------- init: smax=-inf, denom=0, out=0 ---------------------------
__global__ void init_kernel(float* __restrict__ smax, float* __restrict__ denom,
                            float* __restrict__ out, long long n) {
    long long i = (long long)blockIdx.x * blockDim.x + threadIdx.x;
    if (i < n) {
        smax[i]  = -3.402823466e+38f;
        denom[i] = 0.0f;
        out[i]   = 0.0f;
    }
}

// ---------------- dual projection GEMM via V_WMMA_F32_16X16X4_F32 -----------
// grid.x = ceil(N/16) M-tiles, grid.y = 2 (left / right projection)
// block = 256 threads = 8 waves; wave w owns output columns [16w, 16w+16)
__global__ void proj_gemm_kernel(const float* __restrict__ x,
                                 const float* __restrict__ Wl, const float* __restrict__ bl,
                                 const float* __restrict__ Wr, const float* __restrict__ br,
                                 float* __restrict__ hl, float* __restrict__ hr,
                                 int nrows) {
    const int wave = threadIdx.x >> 5;
    const int lane = threadIdx.x & 31;
    const int half = lane >> 4;            // 0 or 1
    const int l15  = lane & 15;

    const bool right = (blockIdx.y != 0);
    const float* __restrict__ W = right ? Wr : Wl;
    const float* __restrict__ b = right ? br : bl;
    float* __restrict__ h       = right ? hr : hl;

    const int mbase = blockIdx.x * 16;
    int arow = mbase + l15;                // A-matrix row (node index)
    if (arow >= nrows) arow = nrows - 1;   // clamp loads (no divergence)
    const int col  = wave * 16 + l15;      // output feature (B/C/D column)
    const int koff = half * 2;             // K sub-offset within each K=4 step

    const float* __restrict__ arow_p = x + (size_t)arow * DF + koff;
    const float* __restrict__ brow_p = W + (size_t)col  * DF + koff;   // B = W^T

    v8f acc = {};
#pragma unroll 4
    for (int k0 = 0; k0 < DF; k0 += 4) {
        v2f a = *(const v2f*)(arow_p + k0);   // A[m][k0+koff], A[m][k0+koff+1]
        v2f bb = *(const v2f*)(brow_p + k0);  // B[k0+koff][n], B[k0+koff+1][n]
        acc = __builtin_amdgcn_wmma_f32_16x16x4_f32(
            false, a, false, bb, (short)0, acc, false, false);
    }

    const float bias = b[col];
    const int mtop = mbase + half * 8;
    float* __restrict__ hp = h + (size_t)mtop * DF + col;
    if (mbase + 16 <= nrows) {             // full tile: unguarded stores
#pragma unroll
        for (int r = 0; r < 8; ++r)
            hp[(size_t)r * DF] = acc[r] + bias;
    } else {                               // ragged tail tile
#pragma unroll
        for (int r = 0; r < 8; ++r)
            if (mtop + r < nrows)
                hp[(size_t)r * DF] = acc[r] + bias;
    }
}

// ---------------- edge score helper: one wave per edge, float4 per lane -----
// alpha is streamed (410 MB, no reuse): non-temporal load so it does not
// evict the ~128 MB hot node-array working set from the 192 MB L2.
__device__ __forceinline__ v4f edge_score(const float* __restrict__ hl,
                                          const float* __restrict__ hr,
                                          const float* __restrict__ alpha,
                                          long long e, int si, int di, int c) {
    const v4f hlv = *(const v4f*)(hl + (size_t)si * DF + c);
    const v4f hrv = *(const v4f*)(hr + (size_t)di * DF + c);
    const v4f av  = __builtin_nontemporal_load((const v4f*)(alpha + (size_t)e * DF + c));
    v4f s = (hlv + hrv) * av;
#pragma unroll
    for (int i = 0; i < 4; ++i)
        s[i] = s[i] >= 0.f ? s[i] : SLOPE * s[i];
    return s;
}

// ---------------- pass A: segment max over dst ------------------------------
__global__ void edge_max_kernel(const float* __restrict__ hl, const float* __restrict__ hr,
                                const float* __restrict__ alpha,
                                const long long* __restrict__ src,
                                const long long* __restrict__ dst,
                                float* __restrict__ smax, long long E) {
    long long t = (long long)blockIdx.x * blockDim.x + threadIdx.x;
    if (t >= E * 32LL) return;
    const long long e = t >> 5;
    const int c = ((int)t & 31) * 4;
    const int si = (int)src[e], di = (int)dst[e];
    const v4f s = edge_score(hl, hr, alpha, e, si, di, c);
    float* p = smax + (size_t)di * DF + c;
    atom_max_f32(p + 0, s[0]);
    atom_max_f32(p + 1, s[1]);
    atom_max_f32(p + 2, s[2]);
    atom_max_f32(p + 3, s[3]);
}

// ---------------- pass B: segment sum of exp(s - smax[dst]) -----------------
__global__ void edge_denom_kernel(const float* __restrict__ hl, const float* __restrict__ hr,
                                  const float* __restrict__ alpha,
                                  const long long* __restrict__ src,
                                  const long long* __restrict__ dst,
                                  const float* __restrict__ smax,
                                  float* __restrict__ denom, long long E) {
    long long t = (long long)blockIdx.x * blockDim.x + threadIdx.x;
    if (t >= E * 32LL) return;
    const long long e = t >> 5;
    const int c = ((int)t & 31) * 4;
    const int si = (int)src[e], di = (int)dst[e];
    const v4f s = edge_score(hl, hr, alpha, e, si, di, c);
    const v4f mx = *(const v4f*)(smax + (size_t)di * DF + c);
    float* p = denom + (size_t)di * DF + c;
    atom_add_f32(p + 0, __expf(s[0] - mx[0]));
    atom_add_f32(p + 1, __expf(s[1] - mx[1]));
    atom_add_f32(p + 2, __expf(s[2] - mx[2]));
    atom_add_f32(p + 3, __expf(s[3] - mx[3]));
}

// ---------------- pass C: out[dst] += x[src] * exp(s-mx)/denom[dst] ---------
__global__ void edge_aggregate_kernel(const float* __restrict__ hl, const float* __restrict__ hr,
                                      const float* __restrict__ alpha,
                                      const float* __restrict__ x,
                                      const long long* __restrict__ src,
                                      const long long* __restrict__ dst,
                                      const float* __restrict__ smax,
                                      const float* __restrict__ denom,
                                      float* __restrict__ out, long long E) {
    long long t = (long long)blockIdx.x * blockDim.x + threadIdx.x;
    if (t >= E * 32LL) return;
    const long long e = t >> 5;
    const int c = ((int)t & 31) * 4;
    const int si = (int)src[e], di = (int)dst[e];
    const v4f s = edge_score(hl, hr, alpha, e, si, di, c);
    const v4f mx = *(const v4f*)(smax + (size_t)di * DF + c);
    const v4f dn = *(const v4f*)(denom + (size_t)di * DF + c);
    const v4f xv = *(const v4f*)(x + (size_t)si * DF + c);
    float* p = out + (size_t)di * DF + c;
    atom_add_f32(p + 0, xv[0] * __expf(s[0] - mx[0]) / dn[0]);
    atom_add_f32(p + 1, xv[1] * __expf(s[1] - mx[1]) / dn[1]);
    atom_add_f32(p + 2, xv[2] * __expf(s[2] - mx[2]) / dn[2]);
    atom_add_f32(p + 3, xv[3] * __expf(s[3] - mx[3]) / dn[3]);
}

// ---------------- launch ----------------------------------------------------
extern "C" void kernel_launch(void* const* d_in, const int* in_sizes, int n_in,
                              void* d_out, int out_size, void* d_ws, size_t ws_size,
                              hipStream_t stream) {
    const float*     x     = (const float*)d_in[0];
    const float*     alpha = (const float*)d_in[1];
    const float*     Wl    = (const float*)d_in[2];
    const float*     bl    = (const float*)d_in[3];
    const float*     Wr    = (const float*)d_in[4];
    const float*     br    = (const float*)d_in[5];
    const long long* src   = (const long long*)d_in[6];   // reference dtype int64
    const long long* dst   = (const long long*)d_in[7];
    float*           out   = (float*)d_out;

    const int       N  = in_sizes[0] / DF;
    const long long E  = (long long)in_sizes[6];
    const long long nd = (long long)N * DF;

    float* hl    = (float*)d_ws;          // N*D
    float* hr    = hl + nd;               // N*D
    float* smax  = hr + nd;               // N*D
    float* denom = smax + nd;             // N*D

    // init accumulators / output
    {
        const int blocks = (int)((nd + 255) / 256);
        init_kernel<<<blocks, 256, 0, stream>>>(smax, denom, out, nd);
    }
    // dense projections hl, hr (WMMA f32)
    {
        dim3 grid((N + 15) / 16, 2);
        proj_gemm_kernel<<<grid, 256, 0, stream>>>(x, Wl, bl, Wr, br, hl, hr, N);
    }
    // edge passes: 1 wave per edge
    const long long tot = E * 32LL;
    const int eblocks = (int)((tot + 255) / 256);
    edge_max_kernel<<<eblocks, 256, 0, stream>>>(hl, hr, alpha, src, dst, smax, E);
    edge_denom_kernel<<<eblocks, 256, 0, stream>>>(hl, hr, alpha, src, dst, smax, denom, E);
    edge_aggregate_kernel<<<eblocks, 256, 0, stream>>>(hl, hr, alpha, x, src, dst,
                                                       smax, denom, out, E);
}